// GAT_47339129536791
// MI455X (gfx1250) — compile-verified
//
#include <hip/hip_runtime.h>
#include <math.h>

#define H 128
#define NEG_SLOPE 0.2f
#define EPS_F 1e-16f
#define ASTRIDE 132   // 128 + 4 pad -> conflict-free ds_load_b64 A-fragments

typedef __attribute__((ext_vector_type(2))) float v2f;
typedef __attribute__((ext_vector_type(8))) float v8f;

// ---------- helpers: order-preserving float<->uint for atomic max ----------
__device__ __forceinline__ unsigned f2ord(float f) {
    unsigned u = __float_as_uint(f);
    return (u & 0x80000000u) ? ~u : (u | 0x80000000u);
}
__device__ __forceinline__ float ord2f(unsigned u) {
    return __uint_as_float((u & 0x80000000u) ? (u ^ 0x80000000u) : ~u);
}

__device__ __forceinline__ float apply_act(float v, int act) {
    if (act == 1)      v = v > 0.0f ? v : 0.0f;
    else if (act == 2) v = 1.0f / (1.0f + __expf(-v));
    return v;
}

// ---------------------------------------------------------------------------
// GEMM: out[N,H] = act(in[N,H] @ W[H,H] (+ bias))   via V_WMMA_F32_16X16X4_F32
// Block: 256 threads = 8 waves; block covers 128 rows x 128 cols.
//   - W (64KB) and the 128-row A tile (~66KB, padded stride) staged in LDS
//   - B fragments (invariant across row tiles) preloaded into 64 VGPRs
//   - tail block handled by clamped load addresses (garbage rows only feed
//     unsaved D rows) + guarded stores; full blocks take a uniform branch-free
//     store path (no per-lane exec masking).
// act: 0=none, 1=relu, 2=sigmoid
// ---------------------------------------------------------------------------
__global__ __launch_bounds__(256) void gemm128_wmma(
    const float* __restrict__ in, const float* __restrict__ Wg,
    const float* __restrict__ bias, float* __restrict__ out,
    int N, int act, int has_bias)
{
    __shared__ float Ws[H * H];          // 64 KB weights
    __shared__ float As[128 * ASTRIDE];  // ~66 KB input tile (padded rows)
    const int tid = threadIdx.x;
    const int rowBase = blockIdx.x * 128;

    {   // stage weights: coalesced float4
        const float4* s = (const float4*)Wg;
        float4* d = (float4*)Ws;
        #pragma unroll
        for (int i = 0; i < (H * H / 4) / 256; ++i)
            d[tid + i * 256] = s[tid + i * 256];
    }
    {   // stage A tile: 128 rows x 128 cols, clamped row index for tail block
        #pragma unroll
        for (int i = 0; i < 16; ++i) {
            const int e    = (i * 256 + tid) * 4;   // linear float index
            const int row  = e >> 7;                // 0..127
            const int colf = e & 127;
            int grow = rowBase + row;
            grow = grow < N ? grow : (N - 1);       // clamp, no branch
            const float4 v = *(const float4*)(in + (size_t)grow * H + colf);
            float* dst = &As[row * ASTRIDE + colf];
            dst[0] = v.x; dst[1] = v.y; dst[2] = v.z; dst[3] = v.w;
        }
    }
    __syncthreads();

    const int wave  = tid >> 5;          // 0..7 -> column tile
    const int lane  = tid & 31;
    const int ln15  = lane & 15;
    const int kb    = (lane >> 4) * 2;   // K sub-offset: 0 or 2 (frag layout)
    const int col   = wave * 16;

    // Preload all 32 B fragments (invariant across row tiles): 64 VGPRs
    v2f bfrag[32];
    #pragma unroll
    for (int kk = 0; kk < 32; ++kk) {
        const int k = kk * 4 + kb;
        bfrag[kk].x = Ws[k * H + col + ln15];
        bfrag[kk].y = Ws[(k + 1) * H + col + ln15];
    }

    const int rshift = (lane >> 4) * 8;  // C/D layout: lanes 16-31 -> M+8
    const int ocol = col + ln15;
    const float bv = has_bias ? bias[ocol] : 0.0f;
    const bool fullBlock = (rowBase + 128) <= N;   // uniform across block

    for (int rt = 0; rt < 8; ++rt) {
        const int row0 = rowBase + rt * 16;
        const float* arow = &As[(rt * 16 + ln15) * ASTRIDE + kb];
        v8f c = {};
        #pragma unroll
        for (int kk = 0; kk < 32; ++kk) {
            const v2f a = *(const v2f*)(arow + kk * 4);  // ds_load_2addr_b64 pairs
            c = __builtin_amdgcn_wmma_f32_16x16x4_f32(
                    false, a, false, bfrag[kk], (short)0, c, false, false);
        }
        float* obase = out + (size_t)(row0 + rshift) * H + ocol;
        if (fullBlock) {                 // uniform branch: no exec masking
            #pragma unroll
            for (int r = 0; r < 8; ++r)
                obase[(size_t)r * H] = apply_act(c[r] + bv, act);
        } else {                         // tail block: guarded stores
            #pragma unroll
            for (int r = 0; r < 8; ++r) {
                if (row0 + r + rshift < N)
                    obase[(size_t)r * H] = apply_act(c[r] + bv, act);
            }
        }
    }
}

// ---------------------------------------------------------------------------
// Per-node attention logits: al_s[n] = hW[n] . a_src ; al_d[n] = hW[n] . a_dst
// One wave per node, float4 per lane, wave32 xor-shuffle reduction.
// ---------------------------------------------------------------------------
__global__ __launch_bounds__(256) void node_scores(
    const float* __restrict__ hW, const float* __restrict__ a_src,
    const float* __restrict__ a_dst, float* __restrict__ al_s,
    float* __restrict__ al_d, int N)
{
    const int n = blockIdx.x * 8 + (threadIdx.x >> 5);
    const int lane = threadIdx.x & 31;
    if (n >= N) return;
    const float4 h4 = ((const float4*)(hW + (size_t)n * H))[lane];
    const float4 s4 = ((const float4*)a_src)[lane];
    const float4 d4 = ((const float4*)a_dst)[lane];
    float s = h4.x * s4.x + h4.y * s4.y + h4.z * s4.z + h4.w * s4.w;
    float d = h4.x * d4.x + h4.y * d4.y + h4.z * d4.z + h4.w * d4.w;
    #pragma unroll
    for (int off = 16; off > 0; off >>= 1) {
        s += __shfl_xor(s, off, 32);
        d += __shfl_xor(d, off, 32);
    }
    if (lane == 0) { al_s[n] = s; al_d[n] = d; }
}

// ---------------------------------------------------------------------------
// Per-layer reset: accum=0 over N*H (float4), m_bits=0 (== lowest ordered
// value; always overwritten because every node has a self loop), denom=0.
// ---------------------------------------------------------------------------
__global__ void layer_init(unsigned* __restrict__ m_bits,
                           float* __restrict__ denom,
                           float4* __restrict__ accum4, int N, int NH4)
{
    const int idx = blockIdx.x * 256 + threadIdx.x;
    if (idx < NH4) accum4[idx] = make_float4(0.f, 0.f, 0.f, 0.f);
    if (idx < N) { m_bits[idx] = 0u; denom[idx] = 0.0f; }
}

// Edge pass 1: e = leaky_relu(al_s[src] + al_d[dst]); segment max via atomic.
__global__ void edge_scores(const int* __restrict__ ei, int E0, int ET,
                            const float* __restrict__ al_s,
                            const float* __restrict__ al_d,
                            float* __restrict__ ebuf,
                            unsigned* __restrict__ m_bits)
{
    const int eid = blockIdx.x * 256 + threadIdx.x;
    if (eid >= ET) return;
    int s, d;
    if (eid < E0) { s = ei[eid]; d = ei[E0 + eid]; }
    else          { s = d = eid - E0; }            // implicit self loops
    float e = al_s[s] + al_d[d];
    e = e > 0.0f ? e : NEG_SLOPE * e;
    ebuf[eid] = e;
    atomicMax(&m_bits[d], f2ord(e));
}

// Edge pass 2: ex = exp(e - m[dst]); denom[dst] += ex (L2 float atomics).
__global__ void edge_exp(const int* __restrict__ ei, int E0, int ET,
                         const float* __restrict__ ebuf,
                         const unsigned* __restrict__ m_bits,
                         float* __restrict__ exbuf,
                         float* __restrict__ denom)
{
    const int eid = blockIdx.x * 256 + threadIdx.x;
    if (eid >= ET) return;
    int d;
    if (eid < E0) d = ei[E0 + eid];
    else          d = eid - E0;
    const float ex = __expf(ebuf[eid] - ord2f(m_bits[d]));
    exbuf[eid] = ex;
    atomicAdd(&denom[d], ex);
}

// Edge pass 3: alpha = ex/(denom+eps); accum[dst] += alpha * hW[src].
// One wave per edge; lane l owns the 16B chunk at 16*l, so the gather is a
// single global_load_b128 (512B contiguous per wave, L2-resident) and the
// four atomics per lane hit contiguous addresses.
__global__ __launch_bounds__(256) void edge_aggregate(
    const int* __restrict__ ei, int E0, int ET,
    const float* __restrict__ exbuf, const float* __restrict__ denom,
    const float* __restrict__ hW, float* __restrict__ accum)
{
    const int eid = blockIdx.x * 8 + (threadIdx.x >> 5);
    const int lane = threadIdx.x & 31;
    if (eid >= ET) return;
    int s, d;
    if (eid < E0) { s = ei[eid]; d = ei[E0 + eid]; }
    else          { s = d = eid - E0; }
    const float alpha = exbuf[eid] / (denom[d] + EPS_F);
    const float4 hv = ((const float4*)(hW + (size_t)s * H))[lane];
    float* od = accum + (size_t)d * H + 4 * lane;
    atomicAdd(&od[0], alpha * hv.x);
    atomicAdd(&od[1], alpha * hv.y);
    atomicAdd(&od[2], alpha * hv.z);
    atomicAdd(&od[3], alpha * hv.w);
}

// Node epilogue: h = relu(accum + bias), float4
__global__ void node_epilogue(const float4* __restrict__ accum4,
                              const float4* __restrict__ bias4,
                              float4* __restrict__ out4, int NH4)
{
    const int idx = blockIdx.x * 256 + threadIdx.x;
    if (idx >= NH4) return;
    const float4 a = accum4[idx];
    const float4 b = bias4[idx & (H / 4 - 1)];
    float4 v;
    v.x = fmaxf(a.x + b.x, 0.0f);
    v.y = fmaxf(a.y + b.y, 0.0f);
    v.z = fmaxf(a.z + b.z, 0.0f);
    v.w = fmaxf(a.w + b.w, 0.0f);
    out4[idx] = v;
}

// ---------------------------------------------------------------------------
extern "C" void kernel_launch(void* const* d_in, const int* in_sizes, int n_in,
                              void* d_out, int out_size, void* d_ws, size_t ws_size,
                              hipStream_t stream)
{
    const float* x        = (const float*)d_in[0];
    const int*   ei       = (const int*)  d_in[1];   // [2, E0] row-major
    const float* enc_W    = (const float*)d_in[2];   // [3,H,H]
    const float* enc_b    = (const float*)d_in[3];   // [3,H]
    const float* conv_W   = (const float*)d_in[4];   // [9,H,H]
    const float* att_src  = (const float*)d_in[5];   // [9,H]
    const float* att_dst  = (const float*)d_in[6];   // [9,H]
    const float* conv_b   = (const float*)d_in[7];   // [9,H]
    const float* dec_W    = (const float*)d_in[8];   // [3,H,H]
    const float* dec_b    = (const float*)d_in[9];   // [3,H]
    float* outp = (float*)d_out;

    const int N  = in_sizes[0] / H;      // 50000
    const int E0 = in_sizes[1] / 2;      // 640000
    const int ET = E0 + N;               // + self loops
    const int NH = N * H;
    const int NH4 = NH / 4;

    // ---- carve workspace ----
    char* ws = (char*)d_ws;
    size_t off = 0;
    auto take = [&](size_t bytes) -> char* {
        char* p = ws + off;
        off = (off + bytes + 255) & ~(size_t)255;
        return p;
    };
    float*    A      = (float*)   take((size_t)NH * 4);  // current features
    float*    B      = (float*)   take((size_t)NH * 4);  // h @ W
    float*    C      = (float*)   take((size_t)NH * 4);  // aggregation accum
    float*    al_s   = (float*)   take((size_t)N  * 4);
    float*    al_d   = (float*)   take((size_t)N  * 4);
    unsigned* m_bits = (unsigned*)take((size_t)N  * 4);
    float*    denom  = (float*)   take((size_t)N  * 4);
    float*    ebuf   = (float*)   take((size_t)ET * 4);
    float*    exbuf  = (float*)   take((size_t)ET * 4);
    (void)ws_size;

    const dim3 blk(256);
    const dim3 gGemm((N + 127) / 128);
    const dim3 gNodeW((N + 7) / 8);          // wave-per-node kernels
    const dim3 gNH4((NH4 + 255) / 256);
    const dim3 gEdge((ET + 255) / 256);
    const dim3 gEdgeW((ET + 7) / 8);         // wave-per-edge kernel

    // -------- encoder: ReLU, ReLU, linear --------
    gemm128_wmma<<<gGemm, blk, 0, stream>>>(x, enc_W,             enc_b,       A, N, 1, 1);
    gemm128_wmma<<<gGemm, blk, 0, stream>>>(A, enc_W + 1 * H * H, enc_b + H,   B, N, 1, 1);
    gemm128_wmma<<<gGemm, blk, 0, stream>>>(B, enc_W + 2 * H * H, enc_b + 2*H, A, N, 0, 1);

    // -------- 9 GATConv layers --------
    for (int i = 0; i < 9; ++i) {
        gemm128_wmma<<<gGemm, blk, 0, stream>>>(A, conv_W + (size_t)i * H * H,
                                                nullptr, B, N, 0, 0);
        node_scores<<<gNodeW, blk, 0, stream>>>(B, att_src + i * H, att_dst + i * H,
                                                al_s, al_d, N);
        layer_init<<<gNH4, blk, 0, stream>>>(m_bits, denom, (float4*)C, N, NH4);
        edge_scores<<<gEdge, blk, 0, stream>>>(ei, E0, ET, al_s, al_d, ebuf, m_bits);
        edge_exp<<<gEdge, blk, 0, stream>>>(ei, E0, ET, ebuf, m_bits, exbuf, denom);
        edge_aggregate<<<gEdgeW, blk, 0, stream>>>(ei, E0, ET, exbuf, denom, B, C);
        node_epilogue<<<gNH4, blk, 0, stream>>>((const float4*)C,
                                                (const float4*)(conv_b + i * H),
                                                (float4*)A, NH4);
    }

    // -------- decoder: ReLU, ReLU, sigmoid --------
    gemm128_wmma<<<gGemm, blk, 0, stream>>>(A, dec_W,             dec_b,       B,    N, 1, 1);
    gemm128_wmma<<<gGemm, blk, 0, stream>>>(B, dec_W + 1 * H * H, dec_b + H,   A,    N, 1, 1);
    gemm128_wmma<<<gGemm, blk, 0, stream>>>(A, dec_W + 2 * H * H, dec_b + 2*H, outp, N, 2, 1);
    (void)out_size;
}